// DelayAndSumLinear_9775345566335
// MI455X (gfx1250) — compile-verified
//
#include <hip/hip_runtime.h>

typedef __attribute__((ext_vector_type(2))) float v2f;
typedef __attribute__((ext_vector_type(8))) float v8f;

namespace {
constexpr int kNDet = 128;
constexpr int kNT   = 2048;
constexpr int kP    = 256 * 256;   // NY*NX = 65536 pixels
constexpr int kB    = 4;
constexpr float kW  = 6.283185307179586f / 127.0f;  // 2*pi/(N_DET-1)

__device__ __forceinline__ float hann_w(float d) {
    return 0.5f - 0.5f * __cosf(kW * d);
}
} // namespace

// One wave = 16 pixels. Wave loops over 32 chunks of 4 detectors; per chunk it
// gathers+lerps sino samples (L2-resident, 4 MB) into the WMMA B operand and
// accumulates apod-weighted sums with V_WMMA_F32_16X16X4_F32 (one accumulator
// per batch). A-matrix = Hann weights (staged in LDS) broadcast to all rows.
// Validity is applied as a 0/1 float multiply (like the reference) so the
// gathers stay unconditional and no EXEC-mask branch appears in the loop.
__global__ __launch_bounds__(256)
void das_wmma_kernel(const float* __restrict__ sino,   // (B,1,N_DET,N_T)
                     const float* __restrict__ lut,    // (NY,NX,N_DET,2)
                     float* __restrict__ out)          // (B,1,NY,NX)
{
    __shared__ float s_apod[kNDet];

    const int tid  = threadIdx.x;
    const int lane = tid & 31;
    const int wv   = tid >> 5;
    const int pcol = lane & 15;          // pixel within the wave tile (N index)
    const int kh   = lane >> 4;          // 0: K=0,1   1: K=2,3
    const int p    = blockIdx.x * 128 + wv * 16 + pcol;

    if (tid < kNDet) s_apod[tid] = hann_w((float)tid);
    __syncthreads();

    const float* __restrict__ lrow = lut + (size_t)p * (2 * kNDet); // 1KB/pixel

    v8f acc[kB];
#pragma unroll
    for (int b = 0; b < kB; ++b) acc[b] = (v8f){0,0,0,0,0,0,0,0};

#pragma unroll 4
    for (int d0 = 0; d0 < kNDet; d0 += 4) {
        const int dA = d0 + (kh << 1);   // this lane-half's detector pair
        const int dB = dA + 1;

        // global_prefetch_b8: next LUT chunk of this pixel's row (speculative,
        // harmless past the end of the row).
        __builtin_prefetch((const void*)(lrow + 2 * (dA + 4)), 0, 1);

        // (tofA, alphaA, tofB, alphaB) — 16B aligned (dA is even)
        const float4 lv = *(const float4*)(lrow + 2 * dA);

        const float kfA = floorf(lv.x);
        const float kfB = floorf(lv.z);
        const float mA  = (kfA >= 0.0f && kfA < (float)(kNT - 1)) ? 1.0f : 0.0f;
        const float mB  = (kfB >= 0.0f && kfB < (float)(kNT - 1)) ? 1.0f : 0.0f;
        const int   k0A = (int)fminf(fmaxf(kfA, 0.0f), (float)(kNT - 2));
        const int   k0B = (int)fminf(fmaxf(kfB, 0.0f), (float)(kNT - 2));
        const float aA  = lv.y, aB = lv.w;

        // A-matrix (16x4 f32): identical rows of Hann weights (LDS, b64 read).
        const v2f amat = *(const v2f*)&s_apod[dA];

        const float* sA = sino + (size_t)dA * kNT + k0A;
        const float* sB = sino + (size_t)dB * kNT + k0B;

#pragma unroll
        for (int b = 0; b < kB; ++b) {
            const size_t boff = (size_t)b * (kNDet * kNT);
            // adjacent pair -> compiler merges into one global_load_b64 gather
            const float s0A = sA[boff], s1A = sA[boff + 1];
            const float s0B = sB[boff], s1B = sB[boff + 1];
            v2f bmat;                      // B-matrix (4x16 f32), column = pcol
            bmat.x = (s0A + aA * (s1A - s0A)) * mA;
            bmat.y = (s0B + aB * (s1B - s0B)) * mB;
            acc[b] = __builtin_amdgcn_wmma_f32_16x16x4_f32(
                false, amat, false, bmat, (short)0, acc[b], false, false);
        }
    }

    // Normalization: sum of the Hann window (uniform across lanes; cold code).
    float norm = 0.0f;
#pragma unroll 1
    for (int d = 0; d < kNDet; ++d) norm += hann_w((float)d);
    const float inv = 1.0f / fmaxf(norm, 1.175494351e-38f);

    // All D rows are identical (broadcast A); row 0 lives in acc[b][0] of
    // lanes 0..15, column = lane.
    if (lane < 16) {
#pragma unroll
        for (int b = 0; b < kB; ++b)
            out[(size_t)b * kP + p] = acc[b][0] * inv;
    }
}

extern "C" void kernel_launch(void* const* d_in, const int* in_sizes, int n_in,
                              void* d_out, int out_size, void* d_ws, size_t ws_size,
                              hipStream_t stream) {
    const float* sino = (const float*)d_in[0];  // (4,1,128,2048) f32
    const float* lut  = (const float*)d_in[1];  // (256,256,128,2) f32
    float* out = (float*)d_out;                 // (4,1,256,256) f32
    (void)in_sizes; (void)n_in; (void)out_size; (void)d_ws; (void)ws_size;

    dim3 block(256);                 // 8 waves -> 128 pixels per block
    dim3 grid(kP / 128);             // 512 blocks
    das_wmma_kernel<<<grid, block, 0, stream>>>(sino, lut, out);
}